// Net_32719060861599
// MI455X (gfx1250) — compile-verified
//
#include <hip/hip_runtime.h>

typedef __attribute__((ext_vector_type(2))) float v2f;
typedef __attribute__((ext_vector_type(8))) float v8f;

#define N_NODES 100000
#define N_EDGES 3200000

// ---------------------------------------------------------------------------
// Zero-fill (agg buffers must start at 0 every call; harness does not re-poison)
// ---------------------------------------------------------------------------
__global__ void zero_f32(float* __restrict__ p, int n) {
    int i = blockIdx.x * blockDim.x + threadIdx.x;
    if (i < n) p[i] = 0.0f;
}

// ---------------------------------------------------------------------------
// GEMM1: h1[N,16] = x[N,128] @ W1^T   (W1 is [16,128] row-major)
// One wave32 per 16-row tile, fp32 WMMA 16x16x4 accumulated over K=128.
// A layout: lanes 0-15 -> K = {k0,k0+1}; lanes 16-31 -> K = {k0+2,k0+3}
// B column n comes from W1 row n (B[k][n] = W1[n][k]).
// ---------------------------------------------------------------------------
__global__ void gemm1_wmma(const float* __restrict__ x,
                           const float* __restrict__ W1,
                           float* __restrict__ h1) {
    const int lane = threadIdx.x & 31;
    const int tile = blockIdx.x * (blockDim.x >> 5) + (threadIdx.x >> 5);
    if (tile >= N_NODES / 16) return;           // wave-uniform exit (EXEC stays all-1s for WMMA)

    const int r  = lane & 15;                   // A row within tile == B/D column
    const int kh = lane >> 4;                   // K-pair select (0 or 1)
    const float* xrow = x  + (size_t)(tile * 16 + r) * 128;
    const float* wrow = W1 + (size_t)r * 128;

    v8f acc = {};
#pragma unroll
    for (int k0 = 0; k0 < 128; k0 += 4) {
        const int kk = k0 + 2 * kh;
        v2f a; a.x = xrow[kk]; a.y = xrow[kk + 1];
        v2f b; b.x = wrow[kk]; b.y = wrow[kk + 1];
        acc = __builtin_amdgcn_wmma_f32_16x16x4_f32(false, a, false, b,
                                                    (short)0, acc, false, false);
    }
#pragma unroll
    for (int v = 0; v < 8; ++v) {               // D: VGPR v holds row M = v + 8*kh
        const int m = v + 8 * kh;
        h1[(size_t)(tile * 16 + m) * 16 + r] = acc[v];
    }
}

// ---------------------------------------------------------------------------
// GEMM2 (fused bias+ReLU on input): h2[N,16] = relu(agg1 + b1) @ W2^T
// W2 is [16,16] row-major. K=16 -> 4 WMMA 16x16x4 steps.
// ---------------------------------------------------------------------------
__global__ void gemm2_wmma(const float* __restrict__ agg1,
                           const float* __restrict__ b1,
                           const float* __restrict__ W2,
                           float* __restrict__ h2) {
    const int lane = threadIdx.x & 31;
    const int tile = blockIdx.x * (blockDim.x >> 5) + (threadIdx.x >> 5);
    if (tile >= N_NODES / 16) return;

    const int r  = lane & 15;
    const int kh = lane >> 4;
    const float* in = agg1 + (size_t)(tile * 16 + r) * 16;

    v8f acc = {};
#pragma unroll
    for (int k0 = 0; k0 < 16; k0 += 4) {
        const int kk = k0 + 2 * kh;
        float a0 = in[kk]     + b1[kk];
        float a1 = in[kk + 1] + b1[kk + 1];
        v2f a; a.x = a0 > 0.f ? a0 : 0.f;
               a.y = a1 > 0.f ? a1 : 0.f;
        v2f b; b.x = W2[r * 16 + kk];
               b.y = W2[r * 16 + kk + 1];
        acc = __builtin_amdgcn_wmma_f32_16x16x4_f32(false, a, false, b,
                                                    (short)0, acc, false, false);
    }
#pragma unroll
    for (int v = 0; v < 8; ++v) {
        const int m = v + 8 * kh;
        h2[(size_t)(tile * 16 + m) * 16 + r] = acc[v];
    }
}

// ---------------------------------------------------------------------------
// Edge pass: agg[dst][c] += ew[e] * h[src][c], thread = (edge, channel).
// 16 lanes share one edge: the h-gather is one coalesced 64B line per edge;
// src/dst/ew loads are redundant across the 16 lanes but L0-served.
// Scatter = one global_atomic_add_f32 per thread into the L2-resident agg.
// ---------------------------------------------------------------------------
__global__ void edge_scatter(const int* __restrict__ src,
                             const int* __restrict__ dst,
                             const float* __restrict__ ew,
                             const float* __restrict__ h,
                             float* __restrict__ agg) {
    const long long t = (long long)blockIdx.x * blockDim.x + threadIdx.x;
    const int e = (int)(t >> 4);
    const int c = (int)(t & 15);
    if (e >= N_EDGES) return;
    const int s = src[e];
    const int d = dst[e];
    const float v = ew[e] * h[(size_t)s * 16 + c];
    atomicAdd(&agg[(size_t)d * 16 + c], v);
}

// ---------------------------------------------------------------------------
// Final: per node, relu(agg2+b2) -> logits (Wl [8,16], bl) -> softmax -> out
// ---------------------------------------------------------------------------
__global__ void final_softmax(const float* __restrict__ agg2,
                              const float* __restrict__ b2,
                              const float* __restrict__ Wl,
                              const float* __restrict__ bl,
                              float* __restrict__ out) {
    const int i = blockIdx.x * blockDim.x + threadIdx.x;
    if (i >= N_NODES) return;

    float h[16];
#pragma unroll
    for (int c = 0; c < 16; ++c) {
        const float v = agg2[(size_t)i * 16 + c] + b2[c];
        h[c] = v > 0.f ? v : 0.f;
    }

    float logit[8];
    float mx = -1e30f;
#pragma unroll
    for (int o = 0; o < 8; ++o) {
        float s = bl[o];
#pragma unroll
        for (int c = 0; c < 16; ++c) s += h[c] * Wl[o * 16 + c];
        logit[o] = s;
        mx = fmaxf(mx, s);
    }
    float den = 0.f;
#pragma unroll
    for (int o = 0; o < 8; ++o) { logit[o] = expf(logit[o] - mx); den += logit[o]; }
    const float inv = 1.f / den;
#pragma unroll
    for (int o = 0; o < 8; ++o) out[(size_t)i * 8 + o] = logit[o] * inv;
}

// ---------------------------------------------------------------------------
// Launch. Inputs (setup_inputs order):
//   0:x [N,128] f32   1:edge_index [2,E] int   2:edge_weight [E] f32
//   3:W1 [16,128]     4:b1 [16]   5:W2 [16,16] 6:b2 [16]
//   7:Wl [8,16]       8:bl [8]
// Workspace: h1 | agg1 | h2 ; agg2 aliases h1 (dead after edge pass 1).
// ---------------------------------------------------------------------------
extern "C" void kernel_launch(void* const* d_in, const int* in_sizes, int n_in,
                              void* d_out, int out_size, void* d_ws, size_t ws_size,
                              hipStream_t stream) {
    const float* x  = (const float*)d_in[0];
    const int*   ei = (const int*)d_in[1];
    const float* ew = (const float*)d_in[2];
    const float* W1 = (const float*)d_in[3];
    const float* b1 = (const float*)d_in[4];
    const float* W2 = (const float*)d_in[5];
    const float* b2 = (const float*)d_in[6];
    const float* Wl = (const float*)d_in[7];
    const float* bl = (const float*)d_in[8];
    float* out = (float*)d_out;

    float* ws   = (float*)d_ws;
    float* h1   = ws;                                  // N*16
    float* agg1 = ws + (size_t)N_NODES * 16;           // N*16
    float* h2   = ws + (size_t)N_NODES * 32;           // N*16
    float* agg2 = h1;                                  // alias: h1 dead after edge pass 1

    const int* src = ei;             // edge_index[0,:]
    const int* dst = ei + N_EDGES;   // edge_index[1,:]

    const dim3 blk(256);
    const int gemmGrid = (N_NODES / 16 + 7) / 8;                    // 8 waves/block
    const int zeroGrid = (N_NODES * 16 + 255) / 256;
    const unsigned edgeGrid =
        (unsigned)(((long long)N_EDGES * 16 + 255) / 256);

    hipLaunchKernelGGL(gemm1_wmma,   dim3(gemmGrid), blk, 0, stream, x, W1, h1);
    hipLaunchKernelGGL(zero_f32,     dim3(zeroGrid), blk, 0, stream, agg1, N_NODES * 16);
    hipLaunchKernelGGL(edge_scatter, dim3(edgeGrid), blk, 0, stream, src, dst, ew, h1, agg1);
    hipLaunchKernelGGL(gemm2_wmma,   dim3(gemmGrid), blk, 0, stream, agg1, b1, W2, h2);
    hipLaunchKernelGGL(zero_f32,     dim3(zeroGrid), blk, 0, stream, agg2, N_NODES * 16);
    hipLaunchKernelGGL(edge_scatter, dim3(edgeGrid), blk, 0, stream, src, dst, ew, h2, agg2);
    hipLaunchKernelGGL(final_softmax, dim3((N_NODES + 255) / 256), blk, 0, stream,
                       agg2, b2, Wl, bl, out);
}